// NaiveConv3D_88098369175845
// MI455X (gfx1250) — compile-verified
//
#include <hip/hip_runtime.h>
#include <hip/hip_bf16.h>

typedef float v2f __attribute__((ext_vector_type(2)));
typedef float v8f __attribute__((ext_vector_type(8)));

#define OC      64
#define IC      32
#define IN_D    64
#define IN_HW   64
#define OD      62
#define KVOL    864                 // 32*3*3*3
#define HALFK   432                 // K per staging pass (16 ic)
#define NCH     108                 // 432/4 WMMA-K chunks per pass
#define KS      466                 // sB k-stride: KS%64==18 -> conflict-free ds_load_b64
#define WROWS   66                  // w rows: -2..63 (2 dead rows absorb kw edge stores)
#define NROWS   288                 // 32 ic * 3 kd * 3 kh

// One block per (b, d, h) output row. 256 threads = 8 waves.
// Wave: ocTile = wave>>1 (16 oc), wPair = wave&1 -> two 16-wide w tiles.
// sB is im2col, K-major: sB[(w+2)*KS + k2] = x[b, ic(k), d+kd(k), h+kh(k), w+kw(k)];
// each B fragment (k,k+1) pair is one aligned, bank-conflict-free ds_load_b64.
// Staging: exactly 36 uniform iterations/thread; per thread wx is fixed and the
// row steps by 4, so the sGB read and all 3 kw fan-out stores use immediate
// offsets. Halo wx=64,65 is skipped: it only feeds the discarded w=62,63 cols.
__global__ __launch_bounds__(256)
void conv3d_wmma_f32_kernel(const float* __restrict__ x,
                            const float* __restrict__ wgt,
                            const float* __restrict__ bias,
                            float* __restrict__ out)
{
    __shared__ float sB[WROWS * KS];   // 123,024 B im2col slab (one ic-half)
    __shared__ int   sGB[NROWS];       // global base per (ic,kd,kh) row

    const int bid = blockIdx.x;        // 0 .. 2*62*62-1
    const int b   = bid / (OD * OD);
    const int rem = bid - b * (OD * OD);
    const int d   = rem / OD;
    const int h   = rem - d * OD;

    for (int i = threadIdx.x; i < NROWS; i += 256) {
        int ic = i / 9;
        int r9 = i - ic * 9;
        int kd = r9 / 3;
        int kh = r9 - kd * 3;
        sGB[i] = (((b * IC + ic) * IN_D + (d + kd)) * IN_HW + (h + kh)) * IN_HW;
    }

    const int lane   = threadIdx.x & 31;
    const int half   = lane >> 4;           // K-phase selector
    const int nm     = lane & 15;           // M (A) / N (B,C,D)
    const int wave   = threadIdx.x >> 5;
    const int ocBase = (wave >> 1) * 16;
    const int wBase0 = (wave & 1) * 32;

    // accumulators init with bias (C/D: VGPR r holds M = r + 8*half)
    v8f c0, c1;
    #pragma unroll
    for (int r = 0; r < 8; ++r) {
        float bv = bias[ocBase + r + half * 8];
        c0[r] = bv;
        c1[r] = bv;
    }

    // per-thread staging constants: wx fixed, row base steps by 4 per iteration
    const int wx  = threadIdx.x & 63;
    const int r30 = threadIdx.x >> 6;                   // 0..3
    const float* xw = x + wx;
    float* p0 = sB + (wx + 2) * KS + 3 * r30 - 930;     // >= 2, kw fan-out base

    for (int icHalf = 0; icHalf < 2; ++icHalf) {
        __syncthreads();   // sGB ready (pass 0) / previous GEMM reads done (pass 1)

        // ---- stage: 144 rows x 64 elems, 36 uniform iterations ----
        const int* gbp = sGB + icHalf * 144 + r30;
        #pragma unroll 6
        for (int it = 0; it < 36; ++it) {
            float v  = xw[gbp[4 * it]];                 // ds_load imm + global scale_offset
            float* p = p0 + 12 * it;                    // immediate-offset stores
            p[930] = v;      // kw = 0 -> (k2=3r3,   w=wx)
            p[465] = v;      // kw = 1 -> (k2=3r3+1, w=wx-1)
            p[0]   = v;      // kw = 2 -> (k2=3r3+2, w=wx-2)
        }
        __syncthreads();

        // ---- GEMM over this half: 108 chunks of K=4 ----
        const float* wlane = wgt + (size_t)(ocBase + nm) * KVOL + half * 2 + icHalf * HALFK;
        const float* blane = sB + (wBase0 + nm + 2) * KS + half * 2;

        #pragma unroll 6
        for (int c = 0; c < NCH; ++c) {
            v2f a  = *(const v2f*)(wlane + 4 * c);             // A: K = 4c+2h, 4c+2h+1
            v2f b0 = *(const v2f*)(blane + 4 * c);             // B tile w..w+15
            v2f b1 = *(const v2f*)(blane + 16 * KS + 4 * c);   // B tile w+16..w+31

            c0 = __builtin_amdgcn_wmma_f32_16x16x4_f32(false, a, false, b0, (short)0, c0, false, false);
            c1 = __builtin_amdgcn_wmma_f32_16x16x4_f32(false, a, false, b1, (short)0, c1, false, false);
        }
    }

    // ---- store (guard the two invalid columns w = 62,63) ----
    const int w0 = wBase0 + nm;
    const int w1 = w0 + 16;
    #pragma unroll
    for (int r = 0; r < 8; ++r) {
        int oc = ocBase + r + half * 8;
        size_t base = ((((size_t)b * OC + oc) * OD + d) * OD + h) * OD;
        if (w0 < OD) out[base + w0] = c0[r];
        if (w1 < OD) out[base + w1] = c1[r];
    }
}

extern "C" void kernel_launch(void* const* d_in, const int* in_sizes, int n_in,
                              void* d_out, int out_size, void* d_ws, size_t ws_size,
                              hipStream_t stream) {
    const float* x    = (const float*)d_in[0];
    const float* wgt  = (const float*)d_in[1];
    const float* bias = (const float*)d_in[2];
    float* out        = (float*)d_out;

    dim3 grid(2 * OD * OD);   // one block per (b, d, h)
    dim3 block(256);
    conv3d_wmma_f32_kernel<<<grid, block, 0, stream>>>(x, wgt, bias, out);
}